// MultiHeadSelfAttention_29094108463778
// MI455X (gfx1250) — compile-verified
//
#include <hip/hip_runtime.h>
#include <hip/hip_bf16.h>

// MI455X / gfx1250 fused multi-head self-attention, fp32 via V_WMMA_F32_16X16X4_F32.
// B=4, N=2048, E=64, H=4, D=16. Flash-style fusion: scores never hit HBM.

#define BB 4
#define NN 2048
#define EE 64
#define HH 4
#define DD 16
#define NEGBIG (-1e9f)

typedef __attribute__((ext_vector_type(2))) float v2f;
typedef __attribute__((ext_vector_type(4))) float v4f;
typedef __attribute__((ext_vector_type(8))) float v8f;

// D = A(16x4 f32) * B(4x16 f32) + C(16x16 f32)
// 8-arg form: (neg_a, A, neg_b, B, c_mod, C, reuse_a, reuse_b)
__device__ __forceinline__ v8f wmma4(v2f a, v2f b, v8f c) {
  return __builtin_amdgcn_wmma_f32_16x16x4_f32(
      /*neg_a=*/false, a, /*neg_b=*/false, b,
      /*c_mod=*/(short)0, c, /*reuse_a=*/false, /*reuse_b=*/false);
}

// -------------------------------------------------------------------------
// Kernel 1: QKV projection.
// grid = B*N/16 blocks, 128 threads (4 waves). Wave w computes columns
// [16w,16w+16) of all three projections == head w, d = lane&15.
// Outputs in [B, H, N, D] layout.
// -------------------------------------------------------------------------
__global__ void __launch_bounds__(128) qkv_proj_kernel(
    const float* __restrict__ x,
    const float* __restrict__ Wq, const float* __restrict__ bq,
    const float* __restrict__ Wk, const float* __restrict__ bk,
    const float* __restrict__ Wv, const float* __restrict__ bv,
    float* __restrict__ qo, float* __restrict__ ko, float* __restrict__ vo) {
  __shared__ float sA[16 * EE];  // 16 rows of x, 4 KB, shared by all 4 waves

  const int tile = blockIdx.x;                 // over B*N/16
  const int b    = tile / (NN / 16);
  const int row0 = (tile % (NN / 16)) * 16;
  const int tid  = threadIdx.x;
  const int wave = tid >> 5;
  const int lane = tid & 31;
  const int l16  = lane & 15;
  const int half = lane >> 4;
  const int koff = half * 2;

  // Cooperative load of the 16x64 x-tile (1024 floats, float4-vectorized).
  {
    const float* src = x + ((size_t)b * NN + row0) * EE;
    for (int i = tid; i < 256; i += 128)
      ((v4f*)sA)[i] = ((const v4f*)src)[i];
  }
  __syncthreads();

  const int col = wave * 16 + l16;  // embedding column e = h*16 + d
  const float* Ws[3]  = {Wq, Wk, Wv};
  const float* bs[3]  = {bq, bk, bv};
  float*       outs[3] = {qo, ko, vo};

  for (int m = 0; m < 3; ++m) {
    const float* W = Ws[m];
    const float bias = bs[m][col];
    v8f c;
    for (int i = 0; i < 8; ++i) c[i] = bias;
    // K-loop over E=64 in steps of 4 (16 f32 WMMAs)
    for (int k0 = 0; k0 < EE; k0 += 4) {
      v2f a, bf;
      a[0]  = sA[l16 * EE + k0 + koff];        // A: row=lane&15, K split by half
      a[1]  = sA[l16 * EE + k0 + koff + 1];
      bf[0] = W[(k0 + koff) * EE + col];       // B: K[k][n], n=lane&15
      bf[1] = W[(k0 + koff + 1) * EE + col];
      c = wmma4(a, bf, c);
    }
    // C layout: VGPR r -> row (r + 8*half), col = l16 (= d)
    float* dst = outs[m] + (((size_t)b * HH + wave) * NN + row0) * DD + l16;
    for (int r = 0; r < 8; ++r)
      dst[(size_t)(r + half * 8) * DD] = c[r];
  }
}

// -------------------------------------------------------------------------
// Kernel 2: flash attention over full N per (b,h).
// grid = B*H*(N/128), 256 threads = 8 waves; each wave owns 16 query rows.
// K/V streamed through LDS in 128-key chunks; P round-trips a per-wave
// padded LDS scratch for the C-layout -> A-layout transpose.
// -------------------------------------------------------------------------
__global__ void __launch_bounds__(256) attn_kernel(
    const float* __restrict__ q, const float* __restrict__ k,
    const float* __restrict__ v, const float* __restrict__ mask,
    float* __restrict__ attn /* [B, N, E] */) {
  __shared__ float sK[128 * DD];       // 8 KB
  __shared__ float sV[128 * DD];       // 8 KB
  __shared__ float sP[8][16 * 18];     // per-wave P scratch, stride 18 (8B-aligned pairs)

  const int rowBlocks = NN / 128;      // 16
  const int bh    = blockIdx.x / rowBlocks;   // b*H + h
  const int b     = bh / HH;
  const int h     = bh % HH;
  const int qbase = (blockIdx.x % rowBlocks) * 128;
  const int tid   = threadIdx.x;
  const int wave  = tid >> 5;
  const int lane  = tid & 31;
  const int l16   = lane & 15;
  const int half  = lane >> 4;
  const int koff  = half * 2;
  const float scale = 0.25f;           // 1/sqrt(D), folded into Q

  // Q tile (16 rows) in WMMA-A layout, pre-scaled. 8 floats/lane.
  const int qrow0 = qbase + wave * 16;
  const float* qp = q + ((size_t)bh * NN + qrow0) * DD;
  v2f qa[4];
  for (int t = 0; t < 4; ++t) {
    qa[t][0] = qp[l16 * DD + t * 4 + koff] * scale;
    qa[t][1] = qp[l16 * DD + t * 4 + koff + 1] * scale;
  }

  float m_run[8], l_run[8];
  v8f acc;
  for (int r = 0; r < 8; ++r) { m_run[r] = -3.0e38f; l_run[r] = 0.0f; acc[r] = 0.0f; }

  const float* kp = k + (size_t)bh * NN * DD;
  const float* vp = v + (size_t)bh * NN * DD;
  const float* mp = mask + (size_t)b * NN;
  float* pw = &sP[wave][0];

  for (int ch = 0; ch < NN; ch += 128) {
    __syncthreads();
    // Cooperative load of 128x16 K and V chunks (float4-vectorized).
    for (int i = tid; i < 512; i += 256) {
      ((v4f*)sK)[i] = ((const v4f*)(kp + (size_t)ch * DD))[i];
      ((v4f*)sV)[i] = ((const v4f*)(vp + (size_t)ch * DD))[i];
    }
    __syncthreads();

    for (int j = 0; j < 8; ++j) {       // 8 key tiles of 16 per chunk
      const int kb = j * 16;
      // S = (Q*scale) . K^T  : 4 x wmma f32 16x16x4
      v8f s;
      for (int r = 0; r < 8; ++r) s[r] = 0.0f;
      for (int t = 0; t < 4; ++t) {
        v2f bf;
        bf[0] = sK[(kb + l16) * DD + t * 4 + koff];      // B[k][n] = K[n][k]
        bf[1] = sK[(kb + l16) * DD + t * 4 + koff + 1];
        s = wmma4(qa[t], bf, s);
      }
      // mask term: per key column (lane&15), identical across the 8 rows
      const float mval = mp[ch + kb + l16] * NEGBIG;

      // online softmax update, row r lives in (VGPR r, half)
      for (int r = 0; r < 8; ++r) {
        float sv = s[r] + mval;
        float tmax = sv;
        for (int off = 1; off < 16; off <<= 1)
          tmax = fmaxf(tmax, __shfl_xor(tmax, off, 16));
        const float m_new = fmaxf(m_run[r], tmax);
        const float alpha = __expf(m_run[r] - m_new);
        const float p     = __expf(sv - m_new);
        float rs = p;
        for (int off = 1; off < 16; off <<= 1)
          rs += __shfl_xor(rs, off, 16);
        l_run[r] = l_run[r] * alpha + rs;
        m_run[r] = m_new;
        acc[r] *= alpha;
        pw[(r + half * 8) * 18 + l16] = p;   // C-layout -> row-major scratch
      }
      // acc += P . V_tile (A from scratch, B from sV); same-wave DS is in-order
      for (int t = 0; t < 4; ++t) {
        v2f a, bf;
        a[0]  = pw[l16 * 18 + t * 4 + koff];
        a[1]  = pw[l16 * 18 + t * 4 + koff + 1];
        bf[0] = sV[(kb + t * 4 + koff) * DD + l16];
        bf[1] = sV[(kb + t * 4 + koff + 1) * DD + l16];
        acc = wmma4(a, bf, acc);
      }
    }
  }

  // normalize and store concat-ready: attn[b, n, h*16 + d]
  float* op = attn + ((size_t)b * NN + qrow0) * EE + h * DD + l16;
  for (int r = 0; r < 8; ++r)
    op[(size_t)(r + half * 8) * EE] = acc[r] / l_run[r];
}

// -------------------------------------------------------------------------
// Kernel 3: output projection out = attn @ Wo + bo. Same shape as kernel 1.
// grid = B*N/16, 128 threads (4 waves, one per 16-col tile).
// -------------------------------------------------------------------------
__global__ void __launch_bounds__(128) out_proj_kernel(
    const float* __restrict__ attn, const float* __restrict__ Wo,
    const float* __restrict__ bo, float* __restrict__ out) {
  __shared__ float sA[16 * EE];
  const int row0 = blockIdx.x * 16;   // over B*N
  const int tid  = threadIdx.x;
  const int wave = tid >> 5;
  const int lane = tid & 31;
  const int l16  = lane & 15;
  const int half = lane >> 4;
  const int koff = half * 2;

  {
    const float* src = attn + (size_t)row0 * EE;
    for (int i = tid; i < 256; i += 128)
      ((v4f*)sA)[i] = ((const v4f*)src)[i];
  }
  __syncthreads();

  const int col = wave * 16 + l16;
  v8f c;
  const float bias = bo[col];
  for (int i = 0; i < 8; ++i) c[i] = bias;
  for (int k0 = 0; k0 < EE; k0 += 4) {
    v2f a, bf;
    a[0]  = sA[l16 * EE + k0 + koff];
    a[1]  = sA[l16 * EE + k0 + koff + 1];
    bf[0] = Wo[(k0 + koff) * EE + col];
    bf[1] = Wo[(k0 + koff + 1) * EE + col];
    c = wmma4(a, bf, c);
  }
  float* dst = out + (size_t)row0 * EE + col;
  for (int r = 0; r < 8; ++r)
    dst[(size_t)(r + half * 8) * EE] = c[r];
}

// -------------------------------------------------------------------------
extern "C" void kernel_launch(void* const* d_in, const int* in_sizes, int n_in,
                              void* d_out, int out_size, void* d_ws, size_t ws_size,
                              hipStream_t stream) {
  const float* x    = (const float*)d_in[0];
  const float* mask = (const float*)d_in[1];
  const float* Wq   = (const float*)d_in[2];
  const float* bq   = (const float*)d_in[3];
  const float* Wk   = (const float*)d_in[4];
  const float* bk   = (const float*)d_in[5];
  const float* Wv   = (const float*)d_in[6];
  const float* bv   = (const float*)d_in[7];
  const float* Wo   = (const float*)d_in[8];
  const float* bo   = (const float*)d_in[9];
  float* out = (float*)d_out;

  // Workspace: Q, K, V in [B,H,N,D] + attn in [B,N,E]; 4 x 2 MiB = 8 MiB.
  const size_t per = (size_t)BB * NN * EE;
  float* qws = (float*)d_ws;
  float* kws = qws + per;
  float* vws = kws + per;
  float* aws = vws + per;

  qkv_proj_kernel<<<BB * NN / 16, 128, 0, stream>>>(
      x, Wq, bq, Wk, bk, Wv, bv, qws, kws, vws);
  attn_kernel<<<BB * HH * (NN / 128), 256, 0, stream>>>(
      qws, kws, vws, mask, aws);
  out_proj_kernel<<<BB * NN / 16, 128, 0, stream>>>(aws, Wo, bo, out);
}